// SeqExtendedContrastiveLoss_3891240370574
// MI455X (gfx1250) — compile-verified
//
#include <hip/hip_runtime.h>
#include <math.h>

typedef __attribute__((ext_vector_type(2))) float v2f;
typedef __attribute__((ext_vector_type(8))) float v8f;

#define B_      16
#define N_      1024
#define D_      256
#define TWO_N   2048
#define TEMP_INV (1.0f / 0.07f)
#define ALPHA_   0.5f
#define TOPK     5
#define INV_2SIG2 0.125f   /* 1/(2*sigma^2), sigma=2 */
#define EPS_     1e-9f
#define NORM_EPS_ 1e-12f

// fp32 WMMA 16x16x4 fragment mapping:
// A frag (16x4): lane -> M = lane%16, K = 2*(lane/16)+v  (v2f)
// B frag (4x16): lane -> N = lane%16, K = 2*(lane/16)+v  (v2f)
// C/D (16x16):   lane,vgpr r -> M = r + 8*(lane/16), N = lane%16 (v8f)
__device__ inline v8f wmma4(v2f a, v2f b, v8f c) {
  return __builtin_amdgcn_wmma_f32_16x16x4_f32(false, a, false, b,
                                               (short)0, c, false, false);
}

// ---------------------------------------------------------------------------
// 1) Row-normalize z1|z2 -> zc [B, 2N, D]; one wave32 per row
// ---------------------------------------------------------------------------
__global__ void normalize_kernel(const float* __restrict__ z1,
                                 const float* __restrict__ z2,
                                 float* __restrict__ zc) {
  const int w = threadIdx.x >> 5, lane = threadIdx.x & 31;
  const int row = blockIdx.x * 8 + w;              // 0 .. B*2N-1
  const int b = row / TWO_N, r = row % TWO_N;
  const float* src = (r < N_) ? (z1 + ((size_t)b * N_ + r) * D_)
                              : (z2 + ((size_t)b * N_ + (r - N_)) * D_);
  float v[8];
  float ss = 0.f;
  for (int t = 0; t < 8; ++t) { v[t] = src[lane + 32 * t]; ss += v[t] * v[t]; }
  for (int m = 16; m >= 1; m >>= 1) ss += __shfl_xor(ss, m, 32);
  const float sc = 1.0f / (sqrtf(ss) + NORM_EPS_);
  float* dst = zc + (size_t)row * D_;
  for (int t = 0; t < 8; ++t) dst[lane + 32 * t] = v[t] * sc;
}

// ---------------------------------------------------------------------------
// 2) A = softmax(zn zn^T) for one half of zc.  Block = (rowTile16, batch).
//    Wave computes 4 col tiles per A-frag (1 LDS + 4 global loads / 4 WMMA).
// ---------------------------------------------------------------------------
__global__ void softmaxA_kernel(const float* __restrict__ zc,
                                float* __restrict__ A, int halfOfs) {
  extern __shared__ float smem[];
  float* sim = smem;                               // 16 * N_
  float* rowsL = smem + 16 * N_;                   // 16 * D_
  const int w = threadIdx.x >> 5, lane = threadIdx.x & 31;
  const int half = lane >> 4, mn = lane & 15;
  const int b = blockIdx.y;
  const int rowBase = blockIdx.x * 16;
  const float* zb = zc + ((size_t)b * TWO_N + halfOfs) * D_;
  for (int idx = threadIdx.x; idx < 16 * D_; idx += 256)
    rowsL[idx] = zb[(size_t)(rowBase + (idx >> 8)) * D_ + (idx & 255)];
  __syncthreads();

  const float* ap = rowsL + mn * D_ + 2 * half;
  for (int cb = w * 64; cb < N_; cb += 512) {
    v8f c[4] = {};
    const float* bp = zb + ((size_t)cb + mn) * D_ + 2 * half;
    for (int k = 0; k < D_; k += 4) {
      v2f a = *(const v2f*)(ap + k);
#pragma unroll
      for (int t = 0; t < 4; ++t) {
        v2f bb = *(const v2f*)(bp + (size_t)t * 16 * D_ + k);
        c[t] = wmma4(a, bb, c[t]);
      }
    }
#pragma unroll
    for (int t = 0; t < 4; ++t)
      for (int r = 0; r < 8; ++r)
        sim[(r + 8 * half) * N_ + cb + t * 16 + mn] = c[t][r];
  }
  __syncthreads();

  for (int rr = w; rr < 16; rr += 8) {             // wave -> rows rr, rr+8
    float* srow = sim + rr * N_;
    float mx = -1e30f;
    for (int j = lane; j < N_; j += 32) mx = fmaxf(mx, srow[j]);
    for (int m = 16; m >= 1; m >>= 1) mx = fmaxf(mx, __shfl_xor(mx, m, 32));
    float s = 0.f;
    for (int j = lane; j < N_; j += 32) {
      float e = expf(srow[j] - mx);
      srow[j] = e;
      s += e;
    }
    for (int m = 16; m >= 1; m >>= 1) s += __shfl_xor(s, m, 32);
    const float inv = 1.0f / s;
    float* arow = A + ((size_t)b * N_ + rowBase + rr) * N_;
    for (int j = lane; j < N_; j += 32) arow[j] = srow[j] * inv;
  }
}

// ---------------------------------------------------------------------------
// 3) C = A @ B per batch ([N x N] row-major), 32x64 register-blocked waves.
//    Optional fused epilogue: C = (addA + addT + A@B) / 3.
// ---------------------------------------------------------------------------
__global__ void gemm_nn32_kernel(const float* __restrict__ A,
                                 const float* __restrict__ Bm,
                                 float* __restrict__ C,
                                 const float* __restrict__ addA,
                                 const float* __restrict__ addT) {
  extern __shared__ float lds[];                   // 32 * N_
  const int w = threadIdx.x >> 5, lane = threadIdx.x & 31;
  const int half = lane >> 4, mn = lane & 15;
  const int b = blockIdx.y;
  const int rowBase = blockIdx.x * 32;
  const size_t bOfs = (size_t)b * N_ * N_;
  const float* Ab = A + bOfs;
  const float* Bb = Bm + bOfs;
  float* Cb = C + bOfs;
  for (int idx = threadIdx.x; idx < 32 * N_; idx += 256)
    lds[idx] = Ab[(size_t)(rowBase + (idx >> 10)) * N_ + (idx & (N_ - 1))];
  __syncthreads();

  const float* ap0 = lds + mn * N_ + 2 * half;
  const float* ap1 = ap0 + 16 * N_;
  for (int it = 0; it < 2; ++it) {
    const int colBase = it * 512 + w * 64;
    const float* bp = Bb + (size_t)(2 * half) * N_ + colBase + mn;
    v8f c0[4] = {};
    v8f c1[4] = {};
    for (int k = 0; k < N_; k += 4) {
      v2f a0 = *(const v2f*)(ap0 + k);
      v2f a1 = *(const v2f*)(ap1 + k);
#pragma unroll
      for (int t = 0; t < 4; ++t) {
        v2f bb;
        bb[0] = bp[(size_t)k * N_ + t * 16];
        bb[1] = bp[(size_t)k * N_ + N_ + t * 16];
        c0[t] = wmma4(a0, bb, c0[t]);
        c1[t] = wmma4(a1, bb, c1[t]);
      }
    }
#pragma unroll
    for (int rt = 0; rt < 2; ++rt)
      for (int t = 0; t < 4; ++t)
        for (int r = 0; r < 8; ++r) {
          const int row = rowBase + rt * 16 + r + 8 * half;
          const int col = colBase + t * 16 + mn;
          const size_t o = (size_t)row * N_ + col;
          float p = rt ? c1[t][r] : c0[t][r];
          if (addA)
            p = (addA[bOfs + o] + addT[bOfs + o] + p) * (1.0f / 3.0f);
          Cb[o] = p;
        }
  }
}

// ---------------------------------------------------------------------------
// 4) den[b, row] = sum_j exp(sim/T) - diag; store cross12 quadrant.
//    Block = (rowTile16 of 2N, batch). Deterministic per-wave partials.
// ---------------------------------------------------------------------------
__global__ void den_cross_kernel(const float* __restrict__ zc,
                                 float* __restrict__ cross12,
                                 float* __restrict__ den) {
  extern __shared__ float ldsm[];
  float* rowsL = ldsm;                             // 16 * D_
  float* denPart = ldsm + 16 * D_;                 // 8 waves * 16 rows
  const int w = threadIdx.x >> 5, lane = threadIdx.x & 31;
  const int half = lane >> 4, mn = lane & 15;
  const int b = blockIdx.y;
  const int rowBase = blockIdx.x * 16;             // 0..2047
  const float* zb = zc + (size_t)b * TWO_N * D_;
  for (int idx = threadIdx.x; idx < 16 * D_; idx += 256)
    rowsL[idx] = zb[(size_t)(rowBase + (idx >> 8)) * D_ + (idx & 255)];
  __syncthreads();

  float acc[8];
  for (int r = 0; r < 8; ++r) acc[r] = 0.f;
  const float* ap = rowsL + mn * D_ + 2 * half;
  for (int cb = w * 64; cb < TWO_N; cb += 512) {
    v8f c[4] = {};
    const float* bp = zb + ((size_t)cb + mn) * D_ + 2 * half;
    for (int k = 0; k < D_; k += 4) {
      v2f a = *(const v2f*)(ap + k);
#pragma unroll
      for (int t = 0; t < 4; ++t) {
        v2f bb = *(const v2f*)(bp + (size_t)t * 16 * D_ + k);
        c[t] = wmma4(a, bb, c[t]);
      }
    }
    const bool storeQ = (rowBase < N_) && (cb >= N_);
#pragma unroll
    for (int t = 0; t < 4; ++t)
      for (int r = 0; r < 8; ++r) {
        const int grow = rowBase + r + 8 * half;
        const int gcol = cb + t * 16 + mn;
        float e = expf(c[t][r] * TEMP_INV);
        if (storeQ)
          cross12[((size_t)b * N_ + grow) * N_ + (gcol - N_)] = e;
        if (grow == gcol) e = 0.f;                 // subtract diagonal
        acc[r] += e;
      }
  }
  for (int r = 0; r < 8; ++r)
    for (int m = 8; m >= 1; m >>= 1)
      acc[r] += __shfl_xor(acc[r], m, 32);
  if (mn == 0)
    for (int r = 0; r < 8; ++r)
      denPart[w * 16 + r + 8 * half] = acc[r];
  __syncthreads();
  if (threadIdx.x < 16) {
    float s = 0.f;
    for (int ww = 0; ww < 8; ++ww) s += denPart[ww * 16 + threadIdx.x];
    den[(size_t)b * TWO_N + rowBase + threadIdx.x] = s;
  }
}

// ---------------------------------------------------------------------------
// 5) Per-row top-5 + numerator + per-row loss. One wave32 per output row.
//    cross21[b,i,j] == cross12[b,j,i] (sim symmetric).
// ---------------------------------------------------------------------------
__global__ void topk_loss_kernel(const float* __restrict__ assoc1,
                                 const float* __restrict__ assoc2,
                                 const float* __restrict__ cross12,
                                 const float* __restrict__ den,
                                 float* __restrict__ loss_rows) {
  __shared__ float sval[8][32 * TOPK];
  __shared__ int   sidx[8][32 * TOPK];
  const int w = threadIdx.x >> 5, lane = threadIdx.x & 31;
  const int g = blockIdx.x * 8 + w;                // 0 .. B*2N-1
  const int b = g / TWO_N, r2 = g % TWO_N;
  const bool firstHalf = r2 < N_;
  const int i = firstHalf ? r2 : (r2 - N_);
  const float* assoc = firstHalf ? (assoc2 + ((size_t)b * N_ + i) * N_)
                                 : (assoc1 + ((size_t)b * N_ + i) * N_);
  float tv[TOPK]; int ti[TOPK];
  for (int k = 0; k < TOPK; ++k) { tv[k] = -1e30f; ti[k] = 1 << 30; }
  for (int j = lane; j < N_; j += 32) {
    const float d = (float)(i - j);
    const float sc = ALPHA_ * expf(-d * d * INV_2SIG2)
                   + (1.0f - ALPHA_) * assoc[j];
    if (sc > tv[TOPK - 1] || (sc == tv[TOPK - 1] && j < ti[TOPK - 1])) {
      int k = TOPK - 1;
      while (k > 0 && (sc > tv[k - 1] || (sc == tv[k - 1] && j < ti[k - 1]))) {
        tv[k] = tv[k - 1]; ti[k] = ti[k - 1]; --k;
      }
      tv[k] = sc; ti[k] = j;
    }
  }
  for (int k = 0; k < TOPK; ++k) {
    sval[w][lane * TOPK + k] = tv[k];
    sidx[w][lane * TOPK + k] = ti[k];
  }
  __syncthreads();
  if (lane == 0) {
    float bv[TOPK]; int bi[TOPK];
    for (int k = 0; k < TOPK; ++k) { bv[k] = -1e30f; bi[k] = 1 << 30; }
    for (int t = 0; t < 32 * TOPK; ++t) {
      const float sc = sval[w][t];
      const int j = sidx[w][t];
      if (sc > bv[TOPK - 1] || (sc == bv[TOPK - 1] && j < bi[TOPK - 1])) {
        int k = TOPK - 1;
        while (k > 0 && (sc > bv[k - 1] || (sc == bv[k - 1] && j < bi[k - 1]))) {
          bv[k] = bv[k - 1]; bi[k] = bi[k - 1]; --k;
        }
        bv[k] = sc; bi[k] = j;
      }
    }
    const float* crow = cross12 + (size_t)b * N_ * N_;
    const float strong = crow[(size_t)i * N_ + i];  // diag of both halves
    float num = strong;
    for (int k = 0; k < TOPK; ++k) {
      const int j = bi[k];
      if (j != i && j < N_) {
        const float gath = firstHalf ? crow[(size_t)i * N_ + j]
                                     : crow[(size_t)j * N_ + i];
        num += bv[k] * gath;
      }
    }
    const float dv = den[(size_t)b * TWO_N + r2];
    loss_rows[g] = -logf(num / (dv + EPS_) + EPS_);
  }
}

// ---------------------------------------------------------------------------
// 6) Deterministic mean of loss_rows -> scalar
// ---------------------------------------------------------------------------
__global__ void reduce_kernel(const float* __restrict__ loss_rows,
                              float* __restrict__ out) {
  __shared__ float s[256];
  float acc = 0.f;
  for (int j = threadIdx.x; j < B_ * TWO_N; j += 256) acc += loss_rows[j];
  s[threadIdx.x] = acc;
  __syncthreads();
  for (int m = 128; m >= 1; m >>= 1) {
    if ((int)threadIdx.x < m) s[threadIdx.x] += s[threadIdx.x + m];
    __syncthreads();
  }
  if (threadIdx.x == 0) out[0] = s[0] / (float)(B_ * TWO_N);  // CL_WEIGHT=1
}

// ---------------------------------------------------------------------------
extern "C" void kernel_launch(void* const* d_in, const int* in_sizes, int n_in,
                              void* d_out, int out_size, void* d_ws,
                              size_t ws_size, hipStream_t stream) {
  const float* z1 = (const float*)d_in[0];
  const float* z2 = (const float*)d_in[1];
  float* ws = (float*)d_ws;

  size_t ofs = 0;
  float* zc     = ws + ofs; ofs += (size_t)B_ * TWO_N * D_;   // 8.4M fl
  float* Abuf   = ws + ofs; ofs += (size_t)B_ * N_ * N_;      // 16.8M fl
  float* Tbuf   = ws + ofs; ofs += (size_t)B_ * N_ * N_;      // also cross12
  float* assoc1 = ws + ofs; ofs += (size_t)B_ * N_ * N_;
  float* assoc2 = ws + ofs; ofs += (size_t)B_ * N_ * N_;
  float* den    = ws + ofs; ofs += (size_t)B_ * TWO_N;
  float* lossr  = ws + ofs; ofs += (size_t)B_ * TWO_N;
  (void)in_sizes; (void)n_in; (void)out_size; (void)ws_size;

  const dim3 blk(256);
  const size_t shSoftmax = (size_t)(16 * N_ + 16 * D_) * sizeof(float); // 80 KB
  const size_t shGemm    = (size_t)32 * N_ * sizeof(float);            // 128 KB
  const size_t shDen     = (size_t)(16 * D_ + 128) * sizeof(float);

  normalize_kernel<<<dim3(B_ * TWO_N / 8), blk, 0, stream>>>(z1, z2, zc);

  // diffusion(z1) -> assoc1 : A, T=A^2, assoc1=(A + T + T@T)/3 (fused)
  softmaxA_kernel<<<dim3(N_ / 16, B_), blk, shSoftmax, stream>>>(zc, Abuf, 0);
  gemm_nn32_kernel<<<dim3(N_ / 32, B_), blk, shGemm, stream>>>(
      Abuf, Abuf, Tbuf, nullptr, nullptr);
  gemm_nn32_kernel<<<dim3(N_ / 32, B_), blk, shGemm, stream>>>(
      Tbuf, Tbuf, assoc1, Abuf, Tbuf);

  // diffusion(z2) -> assoc2
  softmaxA_kernel<<<dim3(N_ / 16, B_), blk, shSoftmax, stream>>>(zc, Abuf, N_);
  gemm_nn32_kernel<<<dim3(N_ / 32, B_), blk, shGemm, stream>>>(
      Abuf, Abuf, Tbuf, nullptr, nullptr);
  gemm_nn32_kernel<<<dim3(N_ / 32, B_), blk, shGemm, stream>>>(
      Tbuf, Tbuf, assoc2, Abuf, Tbuf);

  // den + cross12 (into Tbuf, now free)
  den_cross_kernel<<<dim3(TWO_N / 16, B_), blk, shDen, stream>>>(zc, Tbuf, den);

  // top-k numerator + per-row loss, then deterministic mean
  topk_loss_kernel<<<dim3(B_ * TWO_N / 8), blk, 0, stream>>>(assoc1, assoc2,
                                                             Tbuf, den, lossr);
  reduce_kernel<<<dim3(1), blk, sizeof(float) * 256, stream>>>(
      lossr, (float*)d_out);
}